// Scalar3DHMM_45646912422547
// MI455X (gfx1250) — compile-verified
//
#include <hip/hip_runtime.h>
#include <hip/hip_bf16.h>
#include <math.h>

typedef __attribute__((ext_vector_type(2))) float v2f;
typedef __attribute__((ext_vector_type(8))) float v8f;

#define HMM_N    2048
#define HMM_V    10000
#define HMM_B    16
#define HMM_T    16
#define HMM_L    16
#define HMM_XY   16
#define HMM_ZS   8

// ---------------------------------------------------------------------------
// Kernel 1: per-row logsumexp of emis_params [N, V].  One block per row.
// ---------------------------------------------------------------------------
__global__ void hmm_row_lse_kernel(const float* __restrict__ E,
                                   float* __restrict__ lse) {
    __shared__ float red[256];
    const int r = blockIdx.x;
    const int tid = threadIdx.x;
    const float* row = E + (size_t)r * HMM_V;

    float m = -INFINITY;
    for (int j = tid; j < HMM_V; j += 256) m = fmaxf(m, row[j]);
    red[tid] = m; __syncthreads();
    for (int s = 128; s > 0; s >>= 1) {
        if (tid < s) red[tid] = fmaxf(red[tid], red[tid + s]);
        __syncthreads();
    }
    m = red[0]; __syncthreads();

    float sum = 0.f;
    for (int j = tid; j < HMM_V; j += 256) sum += expf(row[j] - m);
    red[tid] = sum; __syncthreads();
    for (int s = 128; s > 0; s >>= 1) {
        if (tid < s) red[tid] += red[tid + s];
        __syncthreads();
    }
    if (tid == 0) lse[r] = m + logf(red[0]);
}

// ---------------------------------------------------------------------------
// Kernel 2: per-state neighbor list + transition log-softmax (exact port of
// the cumsum/clip/scatter logic in transition_log_p).
// ---------------------------------------------------------------------------
__global__ void hmm_prep_kernel(const float* __restrict__ tp,
                                int* __restrict__ nbr7,
                                float* __restrict__ logp7) {
    const int h = blockIdx.x * blockDim.x + threadIdx.x;
    if (h >= HMM_N) return;
    const int z = h >> 8, rem = h & 255, y = rem >> 4, x = rem & 15;
    const int ox[7] = {0, 1, -1, 0,  0, 0, 0};
    const int oy[7] = {0, 0,  0, 1, -1, 0, 0};
    const int oz[7] = {0, 0,  0, 0,  0, 1, 2};

    float logits[7];
    int   nb[7];
    int   vc = 0;
    for (int k = 0; k < 7; ++k) {
        const int nx = x + ox[k], ny = y + oy[k], nz = z + oz[k];
        const bool valid = (nx >= 0) && (nx < HMM_XY) &&
                           (ny >= 0) && (ny < HMM_XY) &&
                           (nz >= 0) && (nz < HMM_ZS);
        if (valid) {
            logits[k] = tp[h * 7 + vc];   // param_idx = cumsum(valid)-1
            ++vc;
            nb[k] = nx + HMM_XY * (ny + HMM_XY * nz);
        } else {
            logits[k] = -INFINITY;
            nb[k] = h;
        }
    }
    float m = -INFINITY;
    for (int k = 0; k < 7; ++k) m = fmaxf(m, logits[k]);
    float s = 0.f;
    for (int k = 0; k < 7; ++k) s += expf(logits[k] - m);  // exp(-inf)=0
    const float lsev = m + logf(s);
    for (int k = 0; k < 7; ++k) {
        logp7[h * 7 + k] = (logits[k] == -INFINITY) ? -INFINITY
                                                    : logits[k] - lsev;
        nbr7[h * 7 + k] = nb[k];
    }
}

// ---------------------------------------------------------------------------
// Kernel 3: emission via WMMA.  For a fixed t:
//   emis[n, b] = sum_l E[n, tok(b,t,l)] - L * lse[n]
//             = ( G(2048x256) x S(256x16) )[n,b] - L*lse[n]
// where S[m,b] = (m>>4 == b).  Multiplying by exact 0/1 and f32-accumulating
// is numerically identical to a sequential sum.  One wave per 16-row tile.
// Gathers are BRANCHLESS (clamped index + select) so EXEC stays all-ones
// through the whole WMMA loop, as the ISA requires.
// Writes into d_out (later overwritten in place by the recursion kernel).
// ---------------------------------------------------------------------------
__global__ void hmm_emis_wmma_kernel(const float* __restrict__ E,
                                     const int* __restrict__ stories,
                                     const float* __restrict__ lse,
                                     float* __restrict__ emisOut) {
    __shared__ int toks[256];
    const int lane = threadIdx.x;          // 0..31, wave32
    const int t    = blockIdx.y;           // 0..15
    const int n0   = blockIdx.x * 16;      // state tile base

    // tokens for this t: toks[b*16 + l] = stories[b, t, l]
    for (int i = lane; i < 256; i += 32) {
        const int b = i >> 4, l = i & 15;
        toks[i] = stories[(b * HMM_T + t) * HMM_L + l];
    }
    __syncthreads();

    // A-fragment: 16x4 f32, row = lane&15; low lanes K=kk,kk+1; high K=kk+2,kk+3
    const int    rowA   = n0 + (lane & 15);
    const float  lseA   = lse[rowA];
    const size_t rowOff = (size_t)rowA * HMM_V;
    const int    khalf  = (lane >> 4) << 1;   // 0 or 2
    const int    col    = lane & 15;          // B/D column index

    v8f c = {};
    for (int kk = 0; kk < 256; kk += 4) {
        const int k0 = kk + khalf;
        const int t0 = toks[k0];
        const int t1 = toks[k0 + 1];
        // branchless gather: unconditional load from clamped index, then
        // select lseA for pad tokens (-1) -> nets 0 after the L*lse term.
        const int i0 = (t0 >= 0) ? t0 : 0;
        const int i1 = (t1 >= 0) ? t1 : 0;
        const float g0 = E[rowOff + i0];
        const float g1 = E[rowOff + i1];
        v2f a, bb;
        a.x = (t0 >= 0) ? g0 : lseA;
        a.y = (t1 >= 0) ? g1 : lseA;
        const float bv = ((k0 >> 4) == col) ? 1.0f : 0.0f;
        bb.x = bv; bb.y = bv;
        c = __builtin_amdgcn_wmma_f32_16x16x4_f32(
                false, a, false, bb, (short)0, c, false, false);
    }

    // D layout: VGPR r, lanes<16 -> M=r, lanes>=16 -> M=r+8; N = lane&15
    const int hiM = (lane >> 4) << 3;   // 0 or 8
    for (int r = 0; r < 8; ++r) {
        const int n = n0 + r + hiM;
        const float val = c[r] - (float)HMM_L * lse[n];
        emisOut[(size_t)t * (HMM_B * HMM_N) + (size_t)col * HMM_N + n] = val;
    }
}

// ---------------------------------------------------------------------------
// Kernel 4: forward recursion.  One block per batch b, alpha kept in LDS,
// T steps separated by workgroup barriers.  Reads emis from `out` and
// overwrites the same element with alpha (same thread: no hazard).
// ---------------------------------------------------------------------------
__global__ void hmm_forward_kernel(const float* __restrict__ prior,
                                   const int* __restrict__ nbr7,
                                   const float* __restrict__ logp7,
                                   float* __restrict__ out) {
    __shared__ float bufA[HMM_N];
    __shared__ float bufB[HMM_N];
    __shared__ float red[256];
    const int tid = threadIdx.x;      // 256 threads
    const int b   = blockIdx.x;       // 0..15

    // prior logsumexp over N (computed redundantly per block; trivial cost)
    float m = -INFINITY;
    for (int i = tid; i < HMM_N; i += 256) m = fmaxf(m, prior[i]);
    red[tid] = m; __syncthreads();
    for (int s = 128; s > 0; s >>= 1) {
        if (tid < s) red[tid] = fmaxf(red[tid], red[tid + s]);
        __syncthreads();
    }
    m = red[0]; __syncthreads();
    float sum = 0.f;
    for (int i = tid; i < HMM_N; i += 256) sum += expf(prior[i] - m);
    red[tid] = sum; __syncthreads();
    for (int s = 128; s > 0; s >>= 1) {
        if (tid < s) red[tid] += red[tid + s];
        __syncthreads();
    }
    const float plse = m + logf(red[0]);
    __syncthreads();

    float* cur = bufA;
    float* nxt = bufB;
    const size_t base = (size_t)b * HMM_N;

    // t = 0: alpha0 = emis0 + log_softmax(prior)
    for (int i = tid; i < HMM_N; i += 256) {
        const float a0 = out[base + i] + prior[i] - plse;
        cur[i] = a0;
        out[base + i] = a0;
    }
    __syncthreads();

    for (int t = 1; t < HMM_T; ++t) {
        const size_t ob = (size_t)t * (HMM_B * HMM_N) + base;
        for (int i = tid; i < HMM_N; i += 256) {
            float v[7];
            float vmax = -INFINITY;
            for (int k = 0; k < 7; ++k) {
                const float lp = logp7[i * 7 + k];
                const int   j  = nbr7[i * 7 + k];
                const float vv = lp + cur[j];   // -inf + finite = -inf
                v[k] = vv;
                vmax = fmaxf(vmax, vv);         // self edge keeps vmax finite
            }
            float s = 0.f;
            for (int k = 0; k < 7; ++k) s += expf(v[k] - vmax);
            const float a = out[ob + i] + vmax + logf(s);
            nxt[i] = a;
            out[ob + i] = a;
        }
        __syncthreads();
        float* tmp = cur; cur = nxt; nxt = tmp;
    }
}

// ---------------------------------------------------------------------------
extern "C" void kernel_launch(void* const* d_in, const int* in_sizes, int n_in,
                              void* d_out, int out_size, void* d_ws, size_t ws_size,
                              hipStream_t stream) {
    (void)in_sizes; (void)n_in; (void)out_size; (void)ws_size;
    const int*   stories = (const int*)  d_in[0];  // [B,T,L]
    /* d_in[1] = story_length (scalar, == 16) */
    const float* trans   = (const float*)d_in[2];  // [N,7]
    const float* emis    = (const float*)d_in[3];  // [N,V]
    const float* prior   = (const float*)d_in[4];  // [N]
    float* out = (float*)d_out;                    // [T,B,N]

    // workspace layout
    char* ws = (char*)d_ws;
    float* lse   = (float*)(ws);                       //  2048 f  (8 KB)
    int*   nbr7  = (int*)  (ws + 8192);                //  2048*7 i (56 KB)
    float* logp7 = (float*)(ws + 8192 + 57344);        //  2048*7 f (56 KB)

    hmm_row_lse_kernel<<<HMM_N, 256, 0, stream>>>(emis, lse);
    hmm_prep_kernel<<<HMM_N / 256, 256, 0, stream>>>(trans, nbr7, logp7);
    hmm_emis_wmma_kernel<<<dim3(HMM_N / 16, HMM_T), 32, 0, stream>>>(
        emis, stories, lse, out);
    hmm_forward_kernel<<<HMM_B, 256, 0, stream>>>(prior, nbr7, logp7, out);
}